// BlockLinear_30562987278891
// MI455X (gfx1250) — compile-verified
//
#include <hip/hip_runtime.h>

// Block-diagonal linear (1024 independent 3x3 Linears over 16384 rows), f32.
// Memory-bound: 402 MB traffic -> ~17us floor at 23.3 TB/s. Compute mapped onto
// V_WMMA_F32_16X16X4_F32 (exact f32) via 48x48 block-diagonal weight tiles.
// Wave-uniform SADDR addressing (readfirstlane), hoisted per-lane offsets,
// non-temporal hints on the streaming x/out traffic (x+out = 402MB > 192MB L2).

typedef float v2f __attribute__((ext_vector_type(2)));
typedef float v4f __attribute__((ext_vector_type(4)));
typedef float v8f __attribute__((ext_vector_type(8)));

#define NROWS   16384
#define NFEAT   3072
#define GSIZE   16          // components per group
#define GCOLS   48          // = GSIZE*3 columns per group
#define NGROUPS 64          // 1024 components / 16
#define NWAVES  8
#define TILES   4           // 16-row tiles per wave
#define XSTR    52          // padded LDS row stride (dwords): 16B-aligned, bank-conflict-free
#define WSTR    52          // padded Wd column stride (dwords)

__global__ __launch_bounds__(256)
void block_linear_wmma_f32(const float* __restrict__ x,
                           const float* __restrict__ W,
                           const float* __restrict__ bias,
                           float* __restrict__ out) {
  __shared__ float WdT[GCOLS * WSTR];          // block-diag weights, column-major
  __shared__ float Bsh[GCOLS];                 // bias for this group's 48 columns
  __shared__ float Xs[NWAVES][16 * XSTR];      // per-wave X staging (16 x 48, padded)

  const int tid  = threadIdx.x;
  const int g    = blockIdx.x;                 // component group
  const int lane = tid & 31;
  // wave index is wave-uniform: force it scalar so bases become SADDR operands
  const int w    = __builtin_amdgcn_readfirstlane(tid >> 5);
  const int m    = lane & 15;                  // row (A) or column (B/D) within tile
  const int hi   = lane >> 4;                  // lane-half: K/M high bit
  const int koff = hi * 2;

  // ---- build Wd (48x48 block-diagonal), column-major with padded stride ----
  for (int idx = tid; idx < GCOLS * WSTR; idx += 256) WdT[idx] = 0.0f;
  if (tid < GCOLS) Bsh[tid] = bias[g * GCOLS + tid];
  __syncthreads();
  if (tid < GSIZE * 9) {
    const int c = tid / 9, rem = tid % 9, o = rem / 3, i = rem % 3;
    const int row = 3 * c + i, col = 3 * c + o;           // y[col] += x[row]*W[c,o,i]
    WdT[col * WSTR + row] = W[((g * GSIZE + c) * 3 + o) * 3 + i];
  }
  __syncthreads();

  // ---- loop-invariant B fragments: only (kb,j) tiles that touch the diagonal ----
  // j=0 (cols 0-15): kb 0..4 ; j=1 (cols 16-31): kb 3..8 ; j=2 (cols 32-47): kb 7..11
  static constexpr int Ji[16]  = {0,0,0,0,0, 1,1,1,1,1,1, 2,2,2,2,2};
  static constexpr int KBi[16] = {0,1,2,3,4, 3,4,5,6,7,8, 7,8,9,10,11};
  v2f Bt[16];
#pragma unroll
  for (int t = 0; t < 16; ++t) {
    const int col = 16 * Ji[t] + m;            // B lane = N
    const int k   = 4 * KBi[t] + koff;         // B: K = vgpr + 2*half
    Bt[t] = *(const v2f*)&WdT[col * WSTR + k]; // ds_load_b64 (8B aligned)
  }
  float bv[3];
#pragma unroll
  for (int j = 0; j < 3; ++j) bv[j] = Bsh[16 * j + m];

  // ---- hoisted per-lane offsets (loop-invariant across tiles) ----
  int loff[6], lsoff[6];
#pragma unroll
  for (int q = 0; q < 6; ++q) {
    const int f = q * 32 + lane;               // float4 index, 12 per row
    const int r = f / 12, c4 = f % 12;
    loff[q]  = r * NFEAT + c4 * 4;             // global dword offset within tile
    lsoff[q] = r * XSTR + c4 * 4;              // LDS dword offset within slice
  }
  const int so = hi * 8 * NFEAT + m;           // per-lane store offset (rows r / r+8)

  const int rowBlock0 = blockIdx.y * (NWAVES * TILES * 16);

  for (int t = 0; t < TILES; ++t) {
    const int row0 = rowBlock0 + (w * TILES + t) * 16;      // scalar (w is uniform)
    const float* __restrict__ xb = x + (size_t)row0 * NFEAT + g * GCOLS;  // SADDR
    float* __restrict__ ob = out + (size_t)row0 * NFEAT + g * GCOLS;      // SADDR

    // ---- stage X tile (16 rows x 48 f32): 6x global_load_b128 (NT) -> LDS ----
#pragma unroll
    for (int q = 0; q < 6; ++q) {
      const v4f v = __builtin_nontemporal_load((const v4f*)(xb + loff[q]));
      *(v4f*)&Xs[w][lsoff[q]] = v;             // ds_store_b128 (16B aligned)
    }
    __syncthreads();

    // ---- A fragments: lane m row, K = vgpr + 2*half -> ds_load_b64 each ----
    v2f Ax[12];
#pragma unroll
    for (int kb = 0; kb < 12; ++kb)
      Ax[kb] = *(const v2f*)&Xs[w][m * XSTR + kb * 4 + koff];

    // ---- 16 WMMAs: D = A*Wd + D, exact f32 ----
    v8f acc[3] = {};
#pragma unroll
    for (int t2 = 0; t2 < 16; ++t2) {
      acc[Ji[t2]] = __builtin_amdgcn_wmma_f32_16x16x4_f32(
          false, Ax[KBi[t2]], false, Bt[t2], (short)0, acc[Ji[t2]], false, false);
    }

    // ---- store: D native layout coalesced; SADDR + imm offsets, NT policy ----
#pragma unroll
    for (int j = 0; j < 3; ++j) {
#pragma unroll
      for (int r = 0; r < 8; ++r) {
        __builtin_nontemporal_store(acc[j][r] + bv[j],
                                    ob + so + r * NFEAT + 16 * j);
      }
    }
    __syncthreads();
  }
}

extern "C" void kernel_launch(void* const* d_in, const int* in_sizes, int n_in,
                              void* d_out, int out_size, void* d_ws, size_t ws_size,
                              hipStream_t stream) {
  (void)in_sizes; (void)n_in; (void)d_ws; (void)ws_size; (void)out_size;
  const float* x    = (const float*)d_in[0];   // [16384, 3072]
  const float* W    = (const float*)d_in[1];   // [1024, 3, 3]
  const float* bias = (const float*)d_in[2];   // [1024, 3]
  float* out = (float*)d_out;                  // [16384, 3072]

  dim3 grid(NGROUPS, NROWS / (NWAVES * TILES * 16));   // (64, 32)
  dim3 block(256);
  block_linear_wmma_f32<<<grid, block, 0, stream>>>(x, W, bias, out);
}